// GELU63_17566416240668
// MI455X (gfx1250) — compile-verified
//
#include <hip/hip_runtime.h>
#include <hip/hip_bf16.h>

// Problem constants (match reference).
static constexpr int B = 8;
static constexpr int T = 2048;
static constexpr int D = 2048;
static constexpr float EPS = 1e-5f;

typedef unsigned int u32;
typedef u32 u32x4 __attribute__((ext_vector_type(4)));
typedef u32 u32x8 __attribute__((ext_vector_type(8)));

// ---------------------------------------------------------------------------
// CDNA5 async global->LDS copy (16B per lane), ASYNCcnt-tracked.
// Inline asm: portable across ROCm7.2/clang-22 and amdgpu-toolchain (the
// clang builtins disagree on arity). VDST = per-lane LDS byte address,
// VADDR = per-lane 64-bit global address, GV mode ("off").
// ---------------------------------------------------------------------------
__device__ __forceinline__ void async_load_b128(void* lds_dst, const void* gsrc) {
    unsigned lds_addr = (unsigned)(unsigned long long)lds_dst; // low 32b of generic = LDS offset
    asm volatile("global_load_async_to_lds_b128 %0, %1, off"
                 :
                 : "v"(lds_addr), "v"(gsrc)
                 : "memory");
}

__device__ __forceinline__ void wait_asynccnt0() {
    asm volatile("s_wait_asynccnt 0x0" ::: "memory");
}

// ---------------------------------------------------------------------------
// CDNA5 Tensor Data Mover: 1-D tile DMA, global -> LDS, TENSORcnt-tracked.
// Builds a D# (groups 0+1) in SGPRs and issues tensor_load_to_lds.
// n_elems = tile length in f32 elements (must fit 16-bit tile_dim0).
// Issue from ONE wave only (EXEC is ignored by TDM; tile is whole-block data).
// ---------------------------------------------------------------------------
__device__ __forceinline__ void tdm_load_1d_f32(void* lds_dst, const void* gsrc,
                                                u32 n_elems) {
    unsigned long long ga = (unsigned long long)gsrc;
    u32 lds_addr = (u32)(unsigned long long)lds_dst;

    u32x4 g0;
    g0[0] = 1u;                                   // count=1, user descriptor
    g0[1] = lds_addr;                             // lds_addr[31:0]
    g0[2] = (u32)ga;                              // global_addr[31:0]
    g0[3] = (u32)((ga >> 32) & 0x01FFFFFFu)       // global_addr[56:32]
          | (2u << 30);                           // type=2 ("image")

    u32x8 g1;
    g1[0] = 2u << 16;                             // workgroup_mask=0, data_size=4B
    g1[1] = (n_elems & 0xFFFFu) << 16;            // tensor_dim0[15:0]  @ bits 63:48
    g1[2] = ((n_elems >> 16) & 0xFFFFu)           // tensor_dim0[31:16] @ bits 79:64
          | (1u << 16);                           // tensor_dim1=1      @ bits 111:80 (lo)
    g1[3] = (n_elems & 0xFFFFu) << 16;            // tensor_dim1 hi=0; tile_dim0 @127:112
    g1[4] = 1u;                                   // tile_dim1=1; tile_dim2=0
    g1[5] = n_elems;                              // tensor_dim0_stride[31:0]
    g1[6] = 0u;                                   // stride hi, tensor_dim1_stride lo
    g1[7] = 0u;                                   // tensor_dim1_stride hi

    asm volatile("tensor_load_to_lds %0, %1"
                 :
                 : "s"(g0), "s"(g1)
                 : "memory");
}

__device__ __forceinline__ void wait_tensorcnt0() {
#if __has_builtin(__builtin_amdgcn_s_wait_tensorcnt)
    __builtin_amdgcn_s_wait_tensorcnt(0);
#else
    asm volatile("s_wait_tensorcnt 0x0" ::: "memory");
#endif
}

// tanh-approx GELU, identical formula to reference.
__device__ __forceinline__ float gelu_f(float v) {
    const float c = 0.7978845608028654f; // sqrt(2/pi)
    float v3 = v * v * v;
    return 0.5f * v * (1.0f + tanhf(c * (v + 0.044715f * v3)));
}

__device__ __forceinline__ float sigmoid_f(float v) {
    return 1.0f / (1.0f + expf(-v));
}

// ---------------------------------------------------------------------------
// K0: scalar prep. ws[0]=1/max(||ema_out||,1e-12), ws[1]=U, ws[2]=rec_rate,
//     ws[3]=exp(log_tau_cos), ws[4]=alpha.
// ---------------------------------------------------------------------------
__global__ __launch_bounds__(256) void prep_kernel(
    const float* __restrict__ ema_out,
    const float* __restrict__ logit_U,
    const float* __restrict__ log_tau_rec,
    const float* __restrict__ log_tau_cos,
    const float* __restrict__ log_blend,
    float* __restrict__ ws) {
    __shared__ float red[8];
    float ss = 0.0f;
    for (int i = threadIdx.x; i < D; i += 256) {
        float v = ema_out[i];
        ss += v * v;
    }
    #pragma unroll
    for (int off = 16; off > 0; off >>= 1) ss += __shfl_down(ss, off, 32);
    if ((threadIdx.x & 31) == 0) red[threadIdx.x >> 5] = ss;
    __syncthreads();
    if (threadIdx.x == 0) {
        float tot = 0.0f;
        #pragma unroll
        for (int w = 0; w < 8; ++w) tot += red[w];
        ws[0] = 1.0f / fmaxf(sqrtf(tot), 1e-12f);
        ws[1] = sigmoid_f(logit_U[0]);
        float ltr = log_tau_rec[0];
        float sp  = (ltr > 20.0f) ? ltr : log1pf(expf(ltr)); // softplus
        float tau = fmaxf(sp, 0.5f);
        ws[2] = 1.0f - expf(-1.0f / tau);
        ws[3] = expf(log_tau_cos[0]);
        ws[4] = sigmoid_f(log_blend[0]);
    }
}

// ---------------------------------------------------------------------------
// K1: sequential depression scan over T. One thread per (b,d) channel
// (16384 channels = 512 waves = the scan's intrinsic parallelism limit).
// Writes the PRE-update r into d_out (r-trace buffer), coalesced 128B/wave.
// Prefetches 6 rows ahead -> global_prefetch_b8 (guard hoisted out of loop).
// ---------------------------------------------------------------------------
__global__ __launch_bounds__(256) void scan_kernel(
    const float* __restrict__ x,
    const float* __restrict__ ema_resource,
    const float* __restrict__ ws,
    float* __restrict__ rout) {
    const int b = blockIdx.x >> 3;                       // 8 chunks of D per b
    const int d = ((blockIdx.x & 7) << 8) + threadIdx.x; // 0..2047
    const float U   = ws[1];
    const float rec = ws[2];

    float r = ema_resource[d]; // r0 broadcast over b
    const float* xp = x    + (size_t)b * T * D + d;
    float*       rp = rout + (size_t)b * T * D + d;

    int t = 0;
    for (; t < T - 6; ++t) {  // main loop: unconditional prefetch 6 rows ahead
        float xv = xp[(size_t)t * D];
        __builtin_prefetch(xp + (size_t)(t + 6) * D, 0, 1);
        float o = gelu_f(xv);
        float f = tanhf(fabsf(o));          // firing (on stop-grad out)
        rp[(size_t)t * D] = r;              // trace records r BEFORE update
        float used = fminf(U * r * f, 0.99f * r);
        r = fminf(fmaxf(r - used + (1.0f - r) * rec, 0.01f), 1.0f);
    }
    for (; t < T; ++t) {      // epilogue: no prefetch
        float xv = xp[(size_t)t * D];
        float o = gelu_f(xv);
        float f = tanhf(fabsf(o));
        rp[(size_t)t * D] = r;
        float used = fminf(U * r * f, 0.99f * r);
        r = fminf(fmaxf(r - used + (1.0f - r) * rec, 0.01f), 1.0f);
    }
}

// ---------------------------------------------------------------------------
// K2: per-row reductions + finalize, 2 rows (t-values) per block, in place on
// d_out. x-tile and r-tile (16KB each, contiguous) are DMA'd into LDS by the
// Tensor Data Mover (wave 0 issues, TENSORcnt); the 8KB ema vector comes in
// via per-lane async b128 copies (ASYNCcnt). After wait+barrier, reduce
// (sum r, sum out^2, dot(out, ema)) with wave32 shuffles, then overwrite the
// r-trace with out = gelu(x) * r/(mean_r+eps) * cos_gate as float4 stores.
// LDS: 16KB (x) + 16KB (r) + 8KB (ema) + scratch = ~40KB -> ~8 blocks/WGP.
// ---------------------------------------------------------------------------
static constexpr int ROWS = 2;

__global__ __launch_bounds__(256) void finalize_kernel(
    const float* __restrict__ x,
    const float* __restrict__ ema_out,
    const float* __restrict__ ws,
    float* __restrict__ out) {
    __shared__ __align__(16) float sx[ROWS * D];
    __shared__ __align__(16) float sr[ROWS * D];
    __shared__ __align__(16) float sem[D];
    __shared__ float red[8][3];
    __shared__ float rowp[ROWS][2]; // [row] = {inv_mean_r, cos_gate}

    const int bt0 = blockIdx.x * ROWS; // flat (b*T + t0); ROWS divides T
    const float* gx = x   + (size_t)bt0 * D;
    float*      gio = out + (size_t)bt0 * D; // r-trace in, final out (in place)

    // ---- TDM: wave 0 DMAs both 16KB tiles into LDS ----
    if ((threadIdx.x >> 5) == 0) {
        tdm_load_1d_f32(sx, gx,  ROWS * D);
        tdm_load_1d_f32(sr, gio, ROWS * D);
        wait_tensorcnt0();
    }
    // ---- async per-lane copies: 8KB ema vector (512 x b128) ----
    for (int i = threadIdx.x; i < D / 4; i += 256) {
        async_load_b128(&sem[i * 4], ema_out + i * 4);
    }
    wait_asynccnt0();   // this wave's async copies landed in LDS
    __syncthreads();    // TDM tiles + all waves' copies visible workgroup-wide

    // ---- row reductions: 4 waves (128 lanes) per row ----
    const int wave = threadIdx.x >> 5;
    const int lane = threadIdx.x & 31;
    const int row  = wave >> 2;   // 0..1
    const int sub  = wave & 3;    // 0..3
    float s_r = 0.0f, s_o2 = 0.0f, s_oe = 0.0f;
    for (int k = sub * 32 + lane; k < D; k += 128) {
        float o = gelu_f(sx[row * D + k]);
        s_r  += sr[row * D + k];
        s_o2 += o * o;
        s_oe += o * sem[k];
    }
    #pragma unroll
    for (int off = 16; off > 0; off >>= 1) {
        s_r  += __shfl_down(s_r,  off, 32);
        s_o2 += __shfl_down(s_o2, off, 32);
        s_oe += __shfl_down(s_oe, off, 32);
    }
    if (lane == 0) {
        red[wave][0] = s_r;
        red[wave][1] = s_o2;
        red[wave][2] = s_oe;
    }
    __syncthreads();

    if (threadIdx.x < ROWS) {
        const int rr = threadIdx.x;
        float tr = 0.0f, to2 = 0.0f, toe = 0.0f;
        #pragma unroll
        for (int j = 0; j < 4; ++j) {
            tr  += red[rr * 4 + j][0];
            to2 += red[rr * 4 + j][1];
            toe += red[rr * 4 + j][2];
        }
        const float inv_ema_n = ws[0];
        const float tau_cos   = ws[3];
        const float alpha     = ws[4];
        float inv_mean = 1.0f / (tr * (1.0f / (float)D) + EPS);
        float denom    = fmaxf(sqrtf(to2), 1e-12f);
        float cos_sim  = toe * inv_ema_n / denom;
        float novelty  = expf(-tau_cos * cos_sim);
        float gate     = (1.0f - alpha) + alpha * novelty;
        rowp[rr][0] = inv_mean;
        rowp[rr][1] = gate;
    }
    __syncthreads();

    // ---- finalize: float4 streams, overwrite the r-trace with the output ----
    const float4* sx4 = (const float4*)sx;
    const float4* sr4 = (const float4*)sr;
    float4*       go4 = (float4*)gio;
    for (int i = threadIdx.x; i < ROWS * D / 4; i += 256) {
        const int rw = i / (D / 4);
        const float scale = rowp[rw][0] * rowp[rw][1]; // inv_mean * gate
        float4 xv = sx4[i];
        float4 rv = sr4[i];
        float4 res;
        res.x = gelu_f(xv.x) * rv.x * scale;
        res.y = gelu_f(xv.y) * rv.y * scale;
        res.z = gelu_f(xv.z) * rv.z * scale;
        res.w = gelu_f(xv.w) * rv.w * scale;
        go4[i] = res;
    }
}

// ---------------------------------------------------------------------------
// Host launcher. Inputs (setup_inputs order): x, ema_resource, ema_out,
// logit_U, log_tau_rec, log_tau_cos, log_blend. All fp32. Output fp32 B*T*D.
// ---------------------------------------------------------------------------
extern "C" void kernel_launch(void* const* d_in, const int* in_sizes, int n_in,
                              void* d_out, int out_size, void* d_ws, size_t ws_size,
                              hipStream_t stream) {
    const float* x            = (const float*)d_in[0];
    const float* ema_resource = (const float*)d_in[1];
    const float* ema_out      = (const float*)d_in[2];
    const float* logit_U      = (const float*)d_in[3];
    const float* log_tau_rec  = (const float*)d_in[4];
    const float* log_tau_cos  = (const float*)d_in[5];
    const float* log_blend    = (const float*)d_in[6];
    float* out = (float*)d_out;
    float* ws  = (float*)d_ws; // uses 5 floats

    prep_kernel<<<1, 256, 0, stream>>>(ema_out, logit_U, log_tau_rec,
                                       log_tau_cos, log_blend, ws);
    scan_kernel<<<B * (D / 256), 256, 0, stream>>>(x, ema_resource, ws, out);
    finalize_kernel<<<(B * T) / ROWS, 256, 0, stream>>>(x, ema_out, ws, out);
}